// VMPModule_84164179132858
// MI455X (gfx1250) — compile-verified
//
#include <hip/hip_runtime.h>

// Voxel max-pooling scatter:
//   out(B=4, C=64, H=512, W=512) = segment_max over points of feat(B, C, N),
//   voxel grid zero-initialized, indices from pcds_ind(B, N, 2).
// Strategy: zero-fill output, then one thread per (b, n) point looping over C.
// Atomic max is done as GLOBAL_ATOMIC_MAX_U32 on float bit patterns (exact for
// v > 0; v <= 0 can never beat the 0.0f init, so those atomics are skipped).

#define H_DIM 512
#define W_DIM 512
#define C_DIM 64
#define B_DIM 4
#define HW    (H_DIM * W_DIM)

__global__ __launch_bounds__(256) void vmp_zero_kernel(float4* __restrict__ out, int n4) {
    int i = blockIdx.x * blockDim.x + threadIdx.x;
    if (i < n4) {
        out[i] = make_float4(0.0f, 0.0f, 0.0f, 0.0f);
    }
}

__global__ __launch_bounds__(256) void vmp_scatter_max_kernel(
    const float* __restrict__ feat,          // (B, C, N)
    const int*   __restrict__ ind,           // (B, N, 2)
    unsigned int* __restrict__ out,          // (B, C, H, W) as uint bit patterns
    int N, float scale_h, float scale_w) {
    int gid = blockIdx.x * blockDim.x + threadIdx.x;
    int total = B_DIM * N;
    if (gid >= total) return;

    int b = gid / N;
    int n = gid - b * N;

    // Coalesced b64 load of the (h, w) index pair for this point.
    const int2 hw = *(const int2*)(ind + ((size_t)(b * N + n)) * 2);
    int h = (int)floorf((float)hw.x * scale_h);
    int w = (int)floorf((float)hw.y * scale_w);
    if ((unsigned)h >= (unsigned)H_DIM || (unsigned)w >= (unsigned)W_DIM) return;

    const float*  fp       = feat + ((size_t)b * C_DIM) * (size_t)N + (size_t)n;
    unsigned int* out_base = out  + ((size_t)b * C_DIM) * (size_t)HW
                                  + (size_t)h * W_DIM + (size_t)w;

    // 64 channel values live at stride N; consecutive lanes hold consecutive n,
    // so each iteration's wave load is a contiguous 128B global_load_b32 burst.
    #pragma unroll 8
    for (int c = 0; c < C_DIM; ++c) {
        if ((c & 7) == 0 && (c + 8) < C_DIM) {
            // gfx1250 global_prefetch_b8: pull the next channel plane's line
            // toward L2/WGP$ while the atomic tail of this group drains.
            __builtin_prefetch(fp + (size_t)(c + 8) * (size_t)N, 0, 0);
        }
        float v = fp[(size_t)c * (size_t)N];
        // v <= 0 can never exceed the 0.0f init (reference clamps with max(.,0)),
        // and for v > 0 unsigned ordering of IEEE bits == float ordering.
        if (v > 0.0f) {
            atomicMax(out_base + (size_t)c * (size_t)HW, __float_as_uint(v));
        }
    }
}

extern "C" void kernel_launch(void* const* d_in, const int* in_sizes, int n_in,
                              void* d_out, int out_size, void* d_ws, size_t ws_size,
                              hipStream_t stream) {
    const float* feat = (const float*)d_in[0];   // (B, C, N, 1) float32
    const int*   ind  = (const int*)d_in[1];     // (B, N, 2, 1) int32

    // Derive N from the index tensor: in_sizes[1] == B * N * 2.
    int N = in_sizes[1] / (B_DIM * 2);

    // Kernel 1: zero-init the voxel grid (out_size floats; multiple of 4).
    int n4 = out_size / 4;
    int zb = (n4 + 255) / 256;
    vmp_zero_kernel<<<zb, 256, 0, stream>>>((float4*)d_out, n4);

    // Kernel 2: scatter-max all B*N points.
    int total = B_DIM * N;
    int sb = (total + 255) / 256;
    vmp_scatter_max_kernel<<<sb, 256, 0, stream>>>(
        feat, ind, (unsigned int*)d_out, N, 1.0f, 1.0f);
}